// HebbianMLP_14508399526340
// MI455X (gfx1250) — compile-verified
//
#include <hip/hip_runtime.h>
#include <stdint.h>

// ---------------------------------------------------------------------------
// HebbianMLP on MI455X (gfx1250): f32 WMMA GEMM pipeline, fully async-fed.
//   fwd:  h0x = x@(W0+W*H0); h10 = relu(h0x)@(W1+W*H1); y1 = relu(h10)@(W2+W*H2)
//   atb:  e = .7e + .3 A^T B ; heb = clip(heb + dopa*e)
// All matmul flops: v_wmma_f32_16x16x4_f32.  All steady-state tile fills:
// global_load_async_to_lds_b128, double-buffered, pipelined on ASYNCcnt.
// Weight fusion (w + W*heb) and ReLU are applied at fragment-load time so
// every LDS tile is a verbatim copy.
// ---------------------------------------------------------------------------

#define ALPHA_V 0.3f
#define CLAMP_V 0.3f

typedef __attribute__((ext_vector_type(2))) float v2f;
typedef __attribute__((ext_vector_type(8))) float v8f;

__device__ __forceinline__ v8f wmma_f32_4(v2f a, v2f b, v8f c) {
  // (neg_a, A, neg_b, B, c_mod, C, reuse_a, reuse_b)
  return __builtin_amdgcn_wmma_f32_16x16x4_f32(false, a, false, b, (short)0, c,
                                               false, false);
}

// 16-byte global -> LDS copy via CDNA5 async engine (ASYNCcnt-tracked).
// Flat address of a __shared__ object: addr[31:0] is the LDS byte offset.
__device__ __forceinline__ void async_copy16(const float* g, float* l) {
  unsigned int loff = (unsigned int)(uintptr_t)l;
  asm volatile("global_load_async_to_lds_b128 %0, %1, off"
               :: "v"(loff), "v"((unsigned long long)(uintptr_t)g)
               : "memory");
}

// Wait until at most N async ops remain outstanding (in-order completion:
// after issuing a new batch of N, this guarantees the previous batch landed).
template <int N>
__device__ __forceinline__ void async_wait() {
  asm volatile("s_wait_asynccnt %0" :: "i"(N) : "memory");
}

template <bool RELU>
__device__ __forceinline__ float maybe_relu(float x) {
  if constexpr (RELU) {
    float r;
    asm("v_max_num_f32 %0, 0, %1" : "=v"(r) : "v"(x));
    return r;
  } else {
    return x;
  }
}

#define BM 128
#define BN 128
#define BK 32
#define AS_STRIDE 36    // (BK+4) floats; 144B row stride (16B multiple)
#define BS_STRIDE 132   // (BN+4) floats; 528B row stride (16B multiple)

// ---------------------------------------------------------------------------
// Forward GEMM: C[M x N] = act(A) @ (Wb + Wv*Wh),  A:[M x K] lda=K (2048),
// weights row-major stride ldb. If Yext==null: C stride 2048 for all cols<N.
// Else (last layer): cols<2048 -> C (d_out y), cols 2048..N-1 -> Yext[r*2+..].
// ---------------------------------------------------------------------------
template <bool RELU>
__global__ __launch_bounds__(256)
void fwd_gemm(const float* __restrict__ A, const float* __restrict__ Wb,
              const float* __restrict__ Wh, const float* __restrict__ Wscal,
              float* __restrict__ C, float* __restrict__ Yext,
              int K, int N, int ldb)
{
  __shared__ __align__(16) float As[2][BM * AS_STRIDE];   // [m][k]
  __shared__ __align__(16) float Wbs[2][BK * BS_STRIDE];  // [k][n] raw base
  __shared__ __align__(16) float Whs[2][BK * BS_STRIDE];  // [k][n] raw heb

  const int t    = threadIdx.x;
  const int lane = t & 31;
  const int wid  = t >> 5;
  const int wm   = (wid & 1) * 64;   // wave row offset in 128x128 tile
  const int wn   = (wid >> 1) * 32;  // wave col offset
  const int nl   = lane & 15;
  const int hl   = lane >> 4;
  const int m0   = blockIdx.y * BM;
  const int n0   = blockIdx.x * BN;
  const bool fullN = (n0 + BN) <= N;
  const float Wv = Wscal[0];

  v8f acc[4][2];
#pragma unroll
  for (int i = 0; i < 4; ++i)
#pragma unroll
    for (int j = 0; j < 2; ++j)
#pragma unroll
      for (int v = 0; v < 8; ++v) acc[i][j][v] = 0.0f;

  const float* Ag  = A + (size_t)m0 * K;
  const float* Wbg = Wb + n0;
  const float* Whg = Wh + n0;

  // ---- async tile issue: A 128x32 (4 b128/thread), W 32x128 (4 b128/thread)
  auto issueA = [&](int k0, int buf) {
#pragma unroll
    for (int q = 0; q < 4; ++q) {
      int c = t + 256 * q;
      int r = c >> 3;                 // 8 chunks per 32-float row
      int off = (c & 7) << 2;
      async_copy16(Ag + (size_t)r * K + k0 + off, &As[buf][r * AS_STRIDE + off]);
    }
  };
  auto issueW = [&](const float* Wg, float* dst, int k0) {
#pragma unroll
    for (int q = 0; q < 4; ++q) {
      int c = t + 256 * q;
      int r = c >> 5;                 // 32 chunks per 128-float row
      int off = (c & 31) << 2;
      async_copy16(Wg + (size_t)(k0 + r) * ldb + off, dst + r * BS_STRIDE + off);
    }
  };

  // ---- compute one K-block from a buffer set (fusion + relu at frag load)
  auto compute = [&](const float* Asb, const float* Wbsb, const float* Whsb) {
#pragma unroll
    for (int kk = 0; kk < BK; kk += 4) {
      const int kb = kk + 2 * hl;     // VGPR v <-> K = v + 2*(lane/16)
      v2f bfrag[2];
#pragma unroll
      for (int j = 0; j < 2; ++j) {
        int n = wn + j * 16 + nl;
        bfrag[j].x = fmaf(Wv, Whsb[kb * BS_STRIDE + n],
                              Wbsb[kb * BS_STRIDE + n]);
        bfrag[j].y = fmaf(Wv, Whsb[(kb + 1) * BS_STRIDE + n],
                              Wbsb[(kb + 1) * BS_STRIDE + n]);
      }
      v2f afrag[4];
#pragma unroll
      for (int i = 0; i < 4; ++i) {
        int m = wm + i * 16 + nl;
        afrag[i].x = maybe_relu<RELU>(Asb[m * AS_STRIDE + kb]);
        afrag[i].y = maybe_relu<RELU>(Asb[m * AS_STRIDE + kb + 1]);
      }
#pragma unroll
      for (int i = 0; i < 4; ++i)
#pragma unroll
        for (int j = 0; j < 2; ++j)
          acc[i][j] = wmma_f32_4(afrag[i], bfrag[j], acc[i][j]);
    }
  };

  if (fullN) {
    // ---- steady state: double-buffered async pipeline (12 copies / batch)
    issueA(0, 0);
    issueW(Wbg, &Wbs[0][0], 0);
    issueW(Whg, &Whs[0][0], 0);
    const int nb = K / BK;
    for (int b = 0; b < nb; ++b) {
      const int cur = b & 1;
      if (b + 1 < nb) {
        const int k0n = (b + 1) * BK;
        issueA(k0n, 1 - cur);
        issueW(Wbg, &Wbs[1 - cur][0], k0n);
        issueW(Whg, &Whs[1 - cur][0], k0n);
        async_wait<12>();             // batch b landed; batch b+1 in flight
      } else {
        async_wait<0>();
      }
      __syncthreads();
      compute(&As[cur][0], &Wbs[cur][0], &Whs[cur][0]);
      __syncthreads();                // protect buffer reuse by batch b+2
    }
  } else {
    // ---- N-edge block (layer 3 cols >= 2048): scalar fused fill, buffer 0.
    for (int q = t; q < BK * BS_STRIDE; q += 256) Whs[0][q] = 0.0f;  // fma no-op
    for (int k0 = 0; k0 < K; k0 += BK) {
      issueA(k0, 0);
#pragma unroll
      for (int q = 0; q < 16; ++q) {
        int c = t + 256 * q;
        int r = c >> 7;
        int off = c & 127;
        int n = n0 + off;
        float v = 0.0f;
        if (n < N) {
          size_t gi = (size_t)(k0 + r) * ldb + n;
          v = fmaf(Wv, Wh[gi], Wb[gi]);
        }
        Wbs[0][r * BS_STRIDE + off] = v;
      }
      async_wait<0>();
      __syncthreads();
      compute(&As[0][0], &Wbs[0][0], &Whs[0][0]);
      __syncthreads();
    }
  }

  // ---- epilogue: C/D layout row = v + 8*(lane/16), col = lane%16
#pragma unroll
  for (int i = 0; i < 4; ++i) {
    const int rbase = m0 + wm + i * 16 + hl * 8;
#pragma unroll
    for (int j = 0; j < 2; ++j) {
      const int c = n0 + wn + j * 16 + nl;
#pragma unroll
      for (int v = 0; v < 8; ++v) {
        const int r = rbase + v;
        float val = acc[i][j][v];
        if (Yext == nullptr) {
          C[(size_t)r * 2048 + c] = val;
        } else if (c < 2048) {
          C[(size_t)r * 2048 + c] = val;
        } else if (c < N) {
          Yext[(size_t)r * 2 + (c - 2048)] = val;
        }
      }
    }
  }
}

// ---------------------------------------------------------------------------
// A^T B + Hebbian update. C[2048 x N] = act(A)^T @ B, K-reduction over rows.
// A:[K x 2048], B:[K x 2048] (full tiles) / Bext:[K x 2] (cols >= 2048).
// K-major LDS tiles serve the transposed A-fragment with row-striped reads
// (no LDS transpose needed). e = .7e + .3 acc ; heb = clip(heb + dopa*e).
// ---------------------------------------------------------------------------
template <bool RELU>
__global__ __launch_bounds__(256)
void atb_update(const float* __restrict__ A, const float* __restrict__ B,
                const float* __restrict__ Bext,
                const float* __restrict__ e_old, const float* __restrict__ heb_old,
                float* __restrict__ e_out, float* __restrict__ heb_out,
                const float* __restrict__ dopa_p,
                int K, int N)
{
  __shared__ __align__(16) float Aks[2][BK * BS_STRIDE];  // [k][i]
  __shared__ __align__(16) float Bks[2][BK * BS_STRIDE];  // [k][j]

  const int t    = threadIdx.x;
  const int lane = t & 31;
  const int wid  = t >> 5;
  const int wm   = (wid & 1) * 64;
  const int wn   = (wid >> 1) * 32;
  const int nl   = lane & 15;
  const int hl   = lane >> 4;
  const int i0   = blockIdx.y * BM;    // output row = A column
  const int n0   = blockIdx.x * BN;    // output col = B column
  const bool fullN = (n0 + BN) <= N;

  v8f acc[4][2];
#pragma unroll
  for (int i = 0; i < 4; ++i)
#pragma unroll
    for (int j = 0; j < 2; ++j)
#pragma unroll
      for (int v = 0; v < 8; ++v) acc[i][j][v] = 0.0f;

  auto issueT = [&](const float* G, int gcol, float* dst, int k0) {
#pragma unroll
    for (int q = 0; q < 4; ++q) {
      int c = t + 256 * q;
      int r = c >> 5;                 // 32 b128 chunks per 128-float row
      int off = (c & 31) << 2;
      async_copy16(G + (size_t)(k0 + r) * 2048 + gcol + off,
                   dst + r * BS_STRIDE + off);
    }
  };

  auto compute = [&](const float* Asb, const float* Bsb) {
#pragma unroll
    for (int kk = 0; kk < BK; kk += 4) {
      const int kb = kk + 2 * hl;
      v2f bfrag[2];
#pragma unroll
      for (int j = 0; j < 2; ++j) {
        int n = wn + j * 16 + nl;
        bfrag[j].x = Bsb[kb * BS_STRIDE + n];
        bfrag[j].y = Bsb[(kb + 1) * BS_STRIDE + n];
      }
      v2f afrag[4];
#pragma unroll
      for (int i = 0; i < 4; ++i) {
        int m = wm + i * 16 + nl;
        afrag[i].x = maybe_relu<RELU>(Asb[kb * BS_STRIDE + m]);
        afrag[i].y = maybe_relu<RELU>(Asb[(kb + 1) * BS_STRIDE + m]);
      }
#pragma unroll
      for (int i = 0; i < 4; ++i)
#pragma unroll
        for (int j = 0; j < 2; ++j)
          acc[i][j] = wmma_f32_4(afrag[i], bfrag[j], acc[i][j]);
    }
  };

  if (fullN) {
    issueT(A, i0, &Aks[0][0], 0);
    issueT(B, n0, &Bks[0][0], 0);
    const int nb = K / BK;
    for (int b = 0; b < nb; ++b) {
      const int cur = b & 1;
      if (b + 1 < nb) {
        const int k0n = (b + 1) * BK;
        issueT(A, i0, &Aks[1 - cur][0], k0n);
        issueT(B, n0, &Bks[1 - cur][0], k0n);
        async_wait<8>();
      } else {
        async_wait<0>();
      }
      __syncthreads();
      compute(&Aks[cur][0], &Bks[cur][0]);
      __syncthreads();
    }
  } else {
    // layer-2 edge tile: cols 2048..2049 live in the 2-wide side buffer
    for (int k0 = 0; k0 < K; k0 += BK) {
      issueT(A, i0, &Aks[0][0], k0);
#pragma unroll
      for (int q = 0; q < 16; ++q) {
        int c = t + 256 * q;
        int r = c >> 7;
        int off = c & 127;
        int n = n0 + off;
        float v = 0.0f;
        if (n < N && Bext != nullptr)
          v = Bext[(size_t)(k0 + r) * 2 + (n - 2048)];
        Bks[0][r * BS_STRIDE + off] = v;
      }
      async_wait<0>();
      __syncthreads();
      compute(&Aks[0][0], &Bks[0][0]);
      __syncthreads();
    }
  }

  const float dopa = dopa_p[0];
#pragma unroll
  for (int i = 0; i < 4; ++i) {
    const int rbase = i0 + wm + i * 16 + hl * 8;
#pragma unroll
    for (int j = 0; j < 2; ++j) {
      const int c = n0 + wn + j * 16 + nl;
      if (c < N) {
#pragma unroll
        for (int v = 0; v < 8; ++v) {
          const int r = rbase + v;
          size_t idx = (size_t)r * N + c;
          float e = fmaf(ALPHA_V, acc[i][j][v], (1.0f - ALPHA_V) * e_old[idx]);
          float hb = fmaf(dopa, e, heb_old[idx]);
          hb = fminf(CLAMP_V, fmaxf(-CLAMP_V, hb));
          e_out[idx]   = e;
          heb_out[idx] = hb;
        }
      }
    }
  }
}

// ---------------------------------------------------------------------------
// dopa = mean tanh(y1[:,2047])  (col 2047 lives in d_out y, stride 2048)
// Wnew = mean tanh(y1[:,2048])  (col 2048 = Yext col 0)
// ---------------------------------------------------------------------------
__global__ __launch_bounds__(256)
void reduce_scalars(const float* __restrict__ y, const float* __restrict__ yext,
                    float* __restrict__ Wnew_out, float* __restrict__ dopa_out,
                    float* __restrict__ dopa_ws)
{
  __shared__ float s1[256], s2[256];
  const int t = threadIdx.x;
  float sd = 0.0f, sw = 0.0f;
  for (int r = t; r < 4096; r += 256) {
    sd += tanhf(y[(size_t)r * 2048 + 2047]);
    sw += tanhf(yext[(size_t)r * 2 + 0]);
  }
  s1[t] = sd; s2[t] = sw;
  __syncthreads();
  for (int s = 128; s > 0; s >>= 1) {
    if (t < s) { s1[t] += s1[t + s]; s2[t] += s2[t + s]; }
    __syncthreads();
  }
  if (t == 0) {
    float d = s1[0] * (1.0f / 4096.0f);
    float w = s2[0] * (1.0f / 4096.0f);
    *Wnew_out = w;
    *dopa_out = d;
    *dopa_ws  = d;
  }
}

// ---------------------------------------------------------------------------
extern "C" void kernel_launch(void* const* d_in, const int* in_sizes, int n_in,
                              void* d_out, int out_size, void* d_ws, size_t ws_size,
                              hipStream_t stream) {
  (void)in_sizes; (void)n_in; (void)out_size; (void)ws_size;

  const float* x     = (const float*)d_in[0];   // 4096x2048
  const float* x2h0  = (const float*)d_in[1];   // 2048x2048
  const float* h02h1 = (const float*)d_in[2];   // 2048x2048
  const float* h12y  = (const float*)d_in[3];   // 2048x2050
  const float* e0i   = (const float*)d_in[4];
  const float* e1i   = (const float*)d_in[5];
  const float* e2i   = (const float*)d_in[6];
  const float* hb0i  = (const float*)d_in[7];
  const float* hb1i  = (const float*)d_in[8];
  const float* hb2i  = (const float*)d_in[9];
  const float* Wsc   = (const float*)d_in[10];  // scalar

  float* out    = (float*)d_out;
  float* y      = out;                              // 4096*2048
  float* Wnew   = y     + (size_t)4096 * 2048;      // 1
  float* dopa_o = Wnew  + 1;                        // 1
  float* e0o    = dopa_o + 1;                       // 2048*2048
  float* e1o    = e0o   + (size_t)2048 * 2048;      // 2048*2048
  float* e2o    = e1o   + (size_t)2048 * 2048;      // 2048*2050
  float* hb0o   = e2o   + (size_t)2048 * 2050;      // 2048*2048
  float* hb1o   = hb0o  + (size_t)2048 * 2048;      // 2048*2048
  float* hb2o   = hb1o  + (size_t)2048 * 2048;      // 2048*2050

  float* ws      = (float*)d_ws;
  float* h0x     = ws;                              // 4096*2048 (pre-relu)
  float* h10     = h0x  + (size_t)4096 * 2048;      // 4096*2048 (pre-relu)
  float* yext    = h10  + (size_t)4096 * 2048;      // 4096*2 (y1 cols 2048/2049)
  float* dopa_ws = yext + (size_t)4096 * 2;         // 1

  dim3 blk(256);

  // h0x = x @ (x2h0 + W*heb0)
  fwd_gemm<false><<<dim3(16, 32), blk, 0, stream>>>(x, x2h0, hb0i, Wsc, h0x,
                                                    nullptr, 2048, 2048, 2048);
  // h10 = relu(h0x) @ (h02h1 + W*heb1)
  fwd_gemm<true><<<dim3(16, 32), blk, 0, stream>>>(h0x, h02h1, hb1i, Wsc, h10,
                                                   nullptr, 2048, 2048, 2048);
  // y1 = relu(h10) @ (h12y + W*heb2): cols<2048 -> d_out y, 2048/2049 -> yext
  fwd_gemm<true><<<dim3(17, 32), blk, 0, stream>>>(h10, h12y, hb2i, Wsc, y,
                                                   yext, 2048, 2050, 2050);
  // dopa / W_new
  reduce_scalars<<<1, blk, 0, stream>>>(y, yext, Wnew, dopa_o, dopa_ws);
  // layer 0: x^T @ h0x
  atb_update<false><<<dim3(16, 16), blk, 0, stream>>>(x, h0x, nullptr, e0i, hb0i,
                                                      e0o, hb0o, dopa_ws, 4096, 2048);
  // layer 1: relu(h0x)^T @ h10
  atb_update<true><<<dim3(16, 16), blk, 0, stream>>>(h0x, h10, nullptr, e1i, hb1i,
                                                     e1o, hb1o, dopa_ws, 4096, 2048);
  // layer 2: relu(h10)^T @ [y | yext]  (N = 2050)
  atb_update<true><<<dim3(17, 16), blk, 0, stream>>>(h10, y, yext, e2i, hb2i,
                                                     e2o, hb2o, dopa_ws, 4096, 2050);
}